// DynamicAttentionMechanism_52029233824339
// MI455X (gfx1250) — compile-verified
//
#include <hip/hip_runtime.h>

#define BATCH 8
#define CIN   64
#define HID   128
#define COUTC 64
#define GH    96
#define GW    96
#define NNODES (GH * GW)          // 9216
#define MROWS  (BATCH * NNODES)   // 73728

typedef __attribute__((ext_vector_type(2))) float v2f;
typedef __attribute__((ext_vector_type(8))) float v8f;

// ---------------------------------------------------------------------------
// dis[n] = 1/sqrt(deg[n]); deg derived analytically from the (quirky) masked
// 8-connected grid: edge s->t exists for shift (dr,dc) iff
// (dr==0 || s.r>0) && (dc==0 || s.c>0), with torus wrap on the index.
// ---------------------------------------------------------------------------
__global__ __launch_bounds__(256) void dis_kernel(float* __restrict__ dis) {
    int n = blockIdx.x * blockDim.x + threadIdx.x;
    if (n >= NNODES) return;
    int r = n / GW, c = n % GW;
    constexpr int DR[8] = {-1, -1, -1, 0, 0, 1, 1, 1};
    constexpr int DC[8] = {-1, 0, 1, -1, 1, -1, 0, 1};
    int deg = 1;  // self loop
#pragma unroll
    for (int d = 0; d < 8; ++d) {
        int sr = r + DR[d]; sr += (sr < 0) ? GH : 0; sr -= (sr >= GH) ? GH : 0;
        int sc = c + DC[d]; sc += (sc < 0) ? GW : 0; sc -= (sc >= GW) ? GW : 0;
        bool ok = (DR[d] == 0 || sr > 0) && (DC[d] == 0 || sc > 0);
        deg += ok ? 1 : 0;
    }
    dis[n] = rsqrtf((float)deg);
}

// ---------------------------------------------------------------------------
// Layer-1 aggregation from the original x layout [B, C, N] (scalar loads:
// +-1 column stencil offsets preclude vector alignment). Output [B, N, C].
// ---------------------------------------------------------------------------
template <int F>
__global__ __launch_bounds__(256) void agg_chw_kernel(const float* __restrict__ in,
                                                      const float* __restrict__ dis,
                                                      float* __restrict__ out) {
    long gid = (long)blockIdx.x * blockDim.x + threadIdx.x;
    int n = (int)(gid % NNODES);
    long t = gid / NNODES;
    int f = (int)(t % F);
    int b = (int)(t / F);
    if (b >= BATCH) return;
    int r = n / GW, c = n % GW;

    const float* chan = in + ((long)b * F + f) * NNODES;
    float dt  = dis[n];
    float acc = dt * chan[n];           // self loop
    constexpr int DR[8] = {-1, -1, -1, 0, 0, 1, 1, 1};
    constexpr int DC[8] = {-1, 0, 1, -1, 1, -1, 0, 1};
#pragma unroll
    for (int d = 0; d < 8; ++d) {
        int sr = r + DR[d]; sr += (sr < 0) ? GH : 0; sr -= (sr >= GH) ? GH : 0;
        int sc = c + DC[d]; sc += (sc < 0) ? GW : 0; sc -= (sc >= GW) ? GW : 0;
        bool ok = (DR[d] == 0 || sr > 0) && (DC[d] == 0 || sc > 0);
        if (ok) {
            int s = sr * GW + sc;
            acc += dis[s] * chan[s];
        }
    }
    out[((long)b * NNODES + n) * (long)F + f] = dt * acc;
}

// ---------------------------------------------------------------------------
// Row-major aggregation [B,N,F] -> [B,N,F], float4-vectorized over features
// (gather base is f-contiguous per node -> every b128 access is 16B aligned).
// ---------------------------------------------------------------------------
template <int F>
__global__ __launch_bounds__(256) void agg_v4_kernel(const float4* __restrict__ in,
                                                     const float* __restrict__ dis,
                                                     float4* __restrict__ out) {
    constexpr int FQ = F / 4;
    long gid = (long)blockIdx.x * blockDim.x + threadIdx.x;
    int fq = (int)(gid % FQ);
    long t = gid / FQ;
    int n = (int)(t % NNODES);
    int b = (int)(t / NNODES);
    if (b >= BATCH) return;
    int r = n / GW, c = n % GW;

    const float4* base = in + (long)b * NNODES * FQ + fq;
    float dt = dis[n];
    float4 v = base[(long)n * FQ];
    float ax = dt * v.x, ay = dt * v.y, az = dt * v.z, aw = dt * v.w;
    constexpr int DR[8] = {-1, -1, -1, 0, 0, 1, 1, 1};
    constexpr int DC[8] = {-1, 0, 1, -1, 1, -1, 0, 1};
#pragma unroll
    for (int d = 0; d < 8; ++d) {
        int sr = r + DR[d]; sr += (sr < 0) ? GH : 0; sr -= (sr >= GH) ? GH : 0;
        int sc = c + DC[d]; sc += (sc < 0) ? GW : 0; sc -= (sc >= GW) ? GW : 0;
        bool ok = (DR[d] == 0 || sr > 0) && (DC[d] == 0 || sc > 0);
        if (ok) {
            int s = sr * GW + sc;
            float ds = dis[s];
            float4 u = base[(long)s * FQ];
            ax = fmaf(ds, u.x, ax); ay = fmaf(ds, u.y, ay);
            az = fmaf(ds, u.z, az); aw = fmaf(ds, u.w, aw);
        }
    }
    float4 o; o.x = dt * ax; o.y = dt * ay; o.z = dt * az; o.w = dt * aw;
    out[((long)b * NNODES + n) * FQ + fq] = o;
}

// ---------------------------------------------------------------------------
// GEMM  out = relu(A[M,K] @ W[K,NC] + bias)  via V_WMMA_F32_16X16X4_F32.
// One wave computes a 16-row x NC strip with NT=NC/16 independent
// accumulator chains: A loaded once per strip (8x less A traffic than
// tile-per-wave) and the independent D->C chains keep the WMMA pipe full.
// W staged whole in LDS with row stride NC+8 so the two half-wave rows
// (k and k+2) land 16 banks apart: no LDS bank conflicts.
// FINAL=true writes transposed into the [B, NC, NNODES] output tensor.
// ---------------------------------------------------------------------------
template <int K, int NC, bool FINAL>
__global__ __launch_bounds__(256) void gemm_bias_relu(const float* __restrict__ A,
                                                      const float* __restrict__ Wm,
                                                      const float* __restrict__ bias,
                                                      float* __restrict__ out) {
    constexpr int NT  = NC / 16;
    constexpr int NCP = NC + 8;                   // 2*NCP = 16 (mod 64) banks
    __shared__ float sW[K * NCP];
    const int tid = threadIdx.x;

    // Cooperative float4 fill of W into padded LDS (row base stays 16B aligned).
    {
        const float4* Wv  = (const float4*)Wm;
        float4*       sWv = (float4*)sW;
#pragma unroll
        for (int i = tid; i < K * NC / 4; i += 256) {
            int row = i / (NC / 4), colq = i % (NC / 4);
            sWv[row * (NCP / 4) + colq] = Wv[i];
        }
    }
    __syncthreads();

    const int wave = tid >> 5, lane = tid & 31;
    const int half = lane >> 4, l16 = lane & 15;
    const long mt = (long)blockIdx.x * 8 + wave;  // 16-row strip index
    if (mt * 16 >= MROWS) return;                 // uniform per wave: EXEC all-ones

    const float* Arow = A + (mt * 16 + l16) * (long)K + 2 * half;
    const float* sB   = sW + (2 * half) * NCP + l16;

    v8f acc[NT];
#pragma unroll
    for (int nt = 0; nt < NT; ++nt) acc[nt] = (v8f){};

#pragma unroll 8
    for (int k = 0; k < K; k += 4) {
        const v2f a = *(const v2f*)(Arow + k);    // 8B-aligned float2
        const float* sBk = sB + k * NCP;
#pragma unroll
        for (int nt = 0; nt < NT; ++nt) {
            v2f bb;
            bb.x = sBk[nt * 16];
            bb.y = sBk[nt * 16 + NCP];
            acc[nt] = __builtin_amdgcn_wmma_f32_16x16x4_f32(
                false, a, false, bb, (short)0, acc[nt], false, false);
        }
    }

#pragma unroll
    for (int nt = 0; nt < NT; ++nt) {
        const int col = nt * 16 + l16;
        const float bv = bias[col];
#pragma unroll
        for (int v = 0; v < 8; ++v) {
            long  orow = mt * 16 + v + 8 * half;
            float val  = acc[nt][v] + bv;
            val = val > 0.f ? val : 0.f;
            if (!FINAL) {
                out[orow * (long)NC + col] = val;
            } else {
                int b = (int)(orow / NNODES), n = (int)(orow % NNODES);
                out[((long)b * NC + col) * (long)NNODES + n] = val;  // [B,Cout,H,W]
            }
        }
    }
}

// ---------------------------------------------------------------------------
extern "C" void kernel_launch(void* const* d_in, const int* in_sizes, int n_in,
                              void* d_out, int out_size, void* d_ws, size_t ws_size,
                              hipStream_t stream) {
    (void)in_sizes; (void)n_in; (void)out_size; (void)ws_size;
    const float* x  = (const float*)d_in[0];
    const float* W1 = (const float*)d_in[1];
    const float* b1 = (const float*)d_in[2];
    const float* W2 = (const float*)d_in[3];
    const float* b2 = (const float*)d_in[4];
    const float* W3 = (const float*)d_in[5];
    const float* b3 = (const float*)d_in[6];
    float* out = (float*)d_out;

    const size_t bufElems = (size_t)MROWS * HID;   // 9,437,184 floats
    float* bufA = (float*)d_ws;                    // aggregated features
    float* bufS = bufA + bufElems;                 // relu(GEMM) output
    float* dis  = bufS + bufElems;                 // NNODES norm values

    dis_kernel<<<(NNODES + 255) / 256, 256, 0, stream>>>(dis);

    const int gemmBlocks = (MROWS / 16) / 8;       // 576: one 16-row strip/wave

    // Layer 1: agg(x^T) [B,N,64] -> relu(@W1 + b1) [B,N,128]
    agg_chw_kernel<CIN>
        <<<(int)(((long)BATCH * CIN * NNODES) / 256), 256, 0, stream>>>(x, dis, bufA);
    gemm_bias_relu<CIN, HID, false>
        <<<gemmBlocks, 256, 0, stream>>>(bufA, W1, b1, bufS);

    // Layer 2
    agg_v4_kernel<HID>
        <<<(int)(((long)BATCH * NNODES * (HID / 4)) / 256), 256, 0, stream>>>(
            (const float4*)bufS, dis, (float4*)bufA);
    gemm_bias_relu<HID, HID, false>
        <<<gemmBlocks, 256, 0, stream>>>(bufA, W2, b2, bufS);

    // Layer 3 (writes transposed [B,Cout,H,W] directly)
    agg_v4_kernel<HID>
        <<<(int)(((long)BATCH * NNODES * (HID / 4)) / 256), 256, 0, stream>>>(
            (const float4*)bufS, dis, (float4*)bufA);
    gemm_bias_relu<HID, COUTC, true>
        <<<gemmBlocks, 256, 0, stream>>>(bufA, W3, b3, out);
}